// Flow_89928025243946
// MI455X (gfx1250) — compile-verified
//
#include <hip/hip_runtime.h>

// ---------------------------------------------------------------------------
// Dopri5 fixed-step integration of x' = A tanh(x)  (n=1024, T=8192), then
// ys = xs @ C^T (256 outputs) via V_WMMA_F32_16X16X4_F32.
//
// Kernel 1: persistent, 64 workgroups x 256 threads. WG w keeps rows
// [16w,16w+16) of A in LDS (row stride 1040 floats -> conflict-free banks).
// Stage vector exchanged through global memory; grid barrier per stage.
// Each thread caches x and k1..k6 (4 elements each) in VGPRs.
//
// Kernel 2: fp32 WMMA GEMM, one wave per 16x16 output tile, K in steps of 4.
// ---------------------------------------------------------------------------

#define NWG     64
#define TPB     256
#define NSTATE  1024
#define NOUT    256
#define ROWS    16          // rows of A owned per workgroup
#define ASTRIDE 1040        // LDS row stride (floats); 1040 % 64 == 16 -> no bank conflicts

typedef float v8f __attribute__((ext_vector_type(8)));
typedef float v2f __attribute__((ext_vector_type(2)));

__device__ __forceinline__ float4 f4_scale(float a, float4 u) {
  float4 r; r.x = a * u.x; r.y = a * u.y; r.z = a * u.z; r.w = a * u.w; return r;
}
__device__ __forceinline__ float4 f4_fma(float a, float4 u, float4 v) {
  v.x = fmaf(a, u.x, v.x); v.y = fmaf(a, u.y, v.y);
  v.z = fmaf(a, u.z, v.z); v.w = fmaf(a, u.w, v.w);
  return v;
}
__device__ __forceinline__ float4 tanh4(float4 v) {
  v.x = tanhf(v.x); v.y = tanhf(v.y); v.z = tanhf(v.z); v.w = tanhf(v.w);
  return v;
}

// Sense-free generation barrier across the whole (persistent) grid.
// cnt/gen live in d_ws and are memset to 0 on every kernel_launch.
__device__ __forceinline__ void grid_barrier(unsigned* cnt, unsigned* gen) {
  __threadfence();          // make this WG's k-writes visible device-wide
  __syncthreads();
  if (threadIdx.x == 0) {
    unsigned g = __hip_atomic_load(gen, __ATOMIC_RELAXED, __HIP_MEMORY_SCOPE_AGENT);
    unsigned prev = __hip_atomic_fetch_add(cnt, 1u, __ATOMIC_ACQ_REL, __HIP_MEMORY_SCOPE_AGENT);
    if (prev == NWG - 1) {
      __hip_atomic_store(cnt, 0u, __ATOMIC_RELAXED, __HIP_MEMORY_SCOPE_AGENT);
      __hip_atomic_fetch_add(gen, 1u, __ATOMIC_RELEASE, __HIP_MEMORY_SCOPE_AGENT);
    } else {
      while (__hip_atomic_load(gen, __ATOMIC_ACQUIRE, __HIP_MEMORY_SCOPE_AGENT) == g)
        __builtin_amdgcn_s_sleep(1);
    }
  }
  __syncthreads();
  __threadfence();          // invalidate stale near-cache lines before k reads
}

// One RK stage: z = tanh(y); k = A z (this WG's 16 rows); barrier; return this
// thread's 4 elements of the freshly assembled global k vector.
__device__ __forceinline__ float4 run_stage(float4 y, float* __restrict__ kout,
                                            const float* Alds, float* zlds,
                                            unsigned* cnt, unsigned* gen,
                                            int wg, int tid) {
  float4 z = tanh4(y);
  *(float4*)(zlds + tid * 4) = z;
  __syncthreads();

  const int r = tid >> 4;           // local row 0..15
  const int l = tid & 15;           // lane within row group
  const float* arow = Alds + r * ASTRIDE;
  float sum = 0.0f;
#pragma unroll 8
  for (int i = 0; i < NSTATE / 16; ++i) {
    const int c = l + 16 * i;
    sum = fmaf(arow[c], zlds[c], sum);
  }
  // reduce 16 lanes of each half-wave group
  sum += __shfl_down(sum, 8, 16);
  sum += __shfl_down(sum, 4, 16);
  sum += __shfl_down(sum, 2, 16);
  sum += __shfl_down(sum, 1, 16);
  if (l == 0) kout[wg * ROWS + r] = sum;

  grid_barrier(cnt, gen);
  return *(const float4*)(kout + tid * 4);
}

__global__ __launch_bounds__(TPB) void dopri5_persistent(
    const float* __restrict__ x0, const float* __restrict__ tbuf,
    const float* __restrict__ A, float* __restrict__ xs,
    float* __restrict__ kb, unsigned* __restrict__ bar, int T) {
  __shared__ float Alds[ROWS * ASTRIDE];
  __shared__ float zlds[NSTATE];
  const int tid = threadIdx.x;
  const int wg  = blockIdx.x;

  // Pin this WG's 16 rows of A into LDS (64 KB payload).
  for (int i = tid; i < ROWS * (NSTATE / 4); i += TPB) {
    const int r  = i >> 8;        // / (NSTATE/4)
    const int c4 = i & 255;
    const float4 v = *(const float4*)(A + (size_t)(wg * ROWS + r) * NSTATE + c4 * 4);
    float* dst = Alds + r * ASTRIDE + c4 * 4;
    dst[0] = v.x; dst[1] = v.y; dst[2] = v.z; dst[3] = v.w;
  }

  const float dt = tbuf[1] - tbuf[0];
  float4 x = *(const float4*)(x0 + tid * 4);
  if (wg == 0) *(float4*)(xs + tid * 4) = x;   // xs[0] = x0
  __syncthreads();

  unsigned* cnt = bar;
  unsigned* gen = bar + 1;
  float* K1 = kb + 0 * NSTATE;
  float* K2 = kb + 1 * NSTATE;
  float* K3 = kb + 2 * NSTATE;
  float* K4 = kb + 3 * NSTATE;
  float* K5 = kb + 4 * NSTATE;
  float* K6 = kb + 5 * NSTATE;

  for (int t = 1; t < T; ++t) {
    float4 k1 = run_stage(x, K1, Alds, zlds, cnt, gen, wg, tid);

    float4 y = f4_fma(dt * (1.0f / 5.0f), k1, x);
    float4 k2 = run_stage(y, K2, Alds, zlds, cnt, gen, wg, tid);

    float4 acc = f4_scale(3.0f / 40.0f, k1);
    acc = f4_fma(9.0f / 40.0f, k2, acc);
    y = f4_fma(dt, acc, x);
    float4 k3 = run_stage(y, K3, Alds, zlds, cnt, gen, wg, tid);

    acc = f4_scale(44.0f / 45.0f, k1);
    acc = f4_fma(-56.0f / 15.0f, k2, acc);
    acc = f4_fma(32.0f / 9.0f, k3, acc);
    y = f4_fma(dt, acc, x);
    float4 k4 = run_stage(y, K4, Alds, zlds, cnt, gen, wg, tid);

    acc = f4_scale(19372.0f / 6561.0f, k1);
    acc = f4_fma(-25360.0f / 2187.0f, k2, acc);
    acc = f4_fma(64448.0f / 6561.0f, k3, acc);
    acc = f4_fma(-212.0f / 729.0f, k4, acc);
    y = f4_fma(dt, acc, x);
    float4 k5 = run_stage(y, K5, Alds, zlds, cnt, gen, wg, tid);

    acc = f4_scale(9017.0f / 3168.0f, k1);
    acc = f4_fma(-355.0f / 33.0f, k2, acc);
    acc = f4_fma(46732.0f / 5247.0f, k3, acc);
    acc = f4_fma(49.0f / 176.0f, k4, acc);
    acc = f4_fma(-5103.0f / 18656.0f, k5, acc);
    y = f4_fma(dt, acc, x);
    float4 k6 = run_stage(y, K6, Alds, zlds, cnt, gen, wg, tid);

    acc = f4_scale(35.0f / 384.0f, k1);
    acc = f4_fma(500.0f / 1113.0f, k3, acc);
    acc = f4_fma(125.0f / 192.0f, k4, acc);
    acc = f4_fma(-2187.0f / 6784.0f, k5, acc);
    acc = f4_fma(11.0f / 84.0f, k6, acc);
    x = f4_fma(dt, acc, x);

    if (wg == 0) *(float4*)(xs + (size_t)t * NSTATE + tid * 4) = x;
  }
}

// ---------------------------------------------------------------------------
// ys[t][o] = sum_s xs[t][s] * C[o][s]   — fp32 WMMA 16x16x4, 1 wave per tile.
// A-frag: lane m (0-15) holds A[m][k0+{0,1}], lane m+16 holds A[m][k0+{2,3}].
// B-frag mirrored (B = C^T, so B[k][n] = C-row n at col k, contiguous float2).
// D layout: VGPR v -> (M = v + 8*half, N = lane&15).
// ---------------------------------------------------------------------------
__global__ __launch_bounds__(TPB) void out_gemm_wmma(
    const float* __restrict__ xs, const float* __restrict__ C,
    float* __restrict__ ys, int T) {
  const int wid  = (int)(blockIdx.x * blockDim.x + threadIdx.x) >> 5;
  const int lane = threadIdx.x & 31;
  const int nTiles = NOUT / 16;                 // 16
  const int mt = wid / nTiles;                  // time tile
  const int nt = wid % nTiles;                  // output tile
  if (mt * 16 >= T) return;

  const int half = lane >> 4;
  const int mn   = lane & 15;
  const float* arow = xs + (size_t)(mt * 16 + mn) * NSTATE;
  const float* brow = C  + (size_t)(nt * 16 + mn) * NSTATE;

  v8f acc = {0.f, 0.f, 0.f, 0.f, 0.f, 0.f, 0.f, 0.f};
  for (int k = 0; k < NSTATE; k += 4) {
    v2f a = *(const v2f*)(arow + k + half * 2);
    v2f b = *(const v2f*)(brow + k + half * 2);
    acc = __builtin_amdgcn_wmma_f32_16x16x4_f32(
        /*neg_a=*/false, a, /*neg_b=*/false, b,
        /*c_mod=*/(short)0, acc, /*reuse_a=*/false, /*reuse_b=*/false);
  }

  const int m0 = mt * 16 + half * 8;
  const int o  = nt * 16 + mn;
#pragma unroll
  for (int v = 0; v < 8; ++v)
    ys[(size_t)(m0 + v) * NOUT + o] = acc[v];
}

extern "C" void kernel_launch(void* const* d_in, const int* in_sizes, int n_in,
                              void* d_out, int out_size, void* d_ws, size_t ws_size,
                              hipStream_t stream) {
  const float* x0 = (const float*)d_in[0];
  const float* t  = (const float*)d_in[1];
  const float* A  = (const float*)d_in[2];
  const float* C  = (const float*)d_in[3];
  const int T = in_sizes[1];                    // 8192

  float* xs = (float*)d_out;                    // [T, 1024]
  float* ys = (float*)d_out + (size_t)T * NSTATE; // [T, 256]

  unsigned* bar = (unsigned*)d_ws;              // [cnt, gen]
  float* kb = (float*)((char*)d_ws + 256);      // 6 * 1024 floats

  // Reset barrier state every launch (graph-capture safe; deterministic).
  hipMemsetAsync(d_ws, 0, 256, stream);

  dopri5_persistent<<<NWG, TPB, 0, stream>>>(x0, t, A, xs, kb, bar, T);

  const int totalWaves = (T / 16) * (NOUT / 16);
  const int blocks = (totalWaves * 32 + TPB - 1) / TPB;
  out_gemm_wmma<<<blocks, TPB, 0, stream>>>(xs, C, ys, T);
}